// Attention_6871947674044
// MI455X (gfx1250) — compile-verified
//
#include <hip/hip_runtime.h>
#include <hip/hip_bf16.h>

// ---------------------------------------------------------------------------
// CDNA5 (gfx1250) implementation of the GLU-recurrent state model.
//   W12 = W_emb2state @ W_input_influence ; b12 = b_emb2state @ W_input_influence
//   U   = emb @ W12 + b12                  (big WMMA GEMM, async-LDS staged)
//   scan: h = LN(GLU((h@Wsc + U[t]) @ Wglu))  (single WG, latency optimized)
//   W34 = W_output_shaper @ W_state2out
//   out = HS @ W34 + b_state2out           (big WMMA GEMM, async-LDS staged)
// All matmuls use V_WMMA_F32_16X16X4_F32 (native f32 matrix pipe, wave32).
// ---------------------------------------------------------------------------

typedef __attribute__((ext_vector_type(2))) float v2f;
typedef __attribute__((ext_vector_type(8))) float v8f;
typedef __attribute__((ext_vector_type(4))) int   v4i;

__device__ __forceinline__ v8f wmma4(v2f a, v2f b, v8f c) {
  return __builtin_amdgcn_wmma_f32_16x16x4_f32(
      /*neg_a=*/false, a, /*neg_b=*/false, b,
      /*c_mod=*/(short)0, c, /*reuse_a=*/false, /*reuse_b=*/false);
}

__device__ __forceinline__ v8f v8f_zero() {
  v8f z = {0.f, 0.f, 0.f, 0.f, 0.f, 0.f, 0.f, 0.f};
  return z;
}

// --- CDNA5 async global->LDS copy (ASYNCcnt-tracked), with sync fallback ----
#if defined(__HIP_DEVICE_COMPILE__) && \
    __has_builtin(__builtin_amdgcn_global_load_async_to_lds_b128)
#define HAVE_ASYNC_LDS 1
#define ASYNC_COPY_B128(gsrc, ldst)                           \
  __builtin_amdgcn_global_load_async_to_lds_b128(             \
      (__attribute__((address_space(1))) v4i*)(gsrc),         \
      (__attribute__((address_space(3))) v4i*)(ldst), 0, 0)
#if __has_builtin(__builtin_amdgcn_s_wait_asynccnt)
#define WAIT_ASYNC(n) __builtin_amdgcn_s_wait_asynccnt(n)
#else
#define WAIT_ASYNC(n) asm volatile("s_wait_asynccnt " #n ::: "memory")
#endif
#endif

// ---------------------------------------------------------------------------
// Generic row-major f32 GEMM: C[M,N] = A[M,K] @ B[K,N] (+ bias[N])
// Block tile 64x128, 256 threads = 8 waves, each wave 32x32 (2x2 WMMA tiles).
// Double-buffered LDS tiles; async direct-to-LDS loads when available.
// Requires M%64==0, N%128==0, K%16==0 (true for all call sites here).
// ---------------------------------------------------------------------------
#define GBM 64
#define GBN 128
#define GBK 16

__global__ __launch_bounds__(256) void gemm_wmma_f32(
    const float* __restrict__ A, const float* __restrict__ B,
    const float* __restrict__ bias, float* __restrict__ C,
    int M, int N, int K) {
  __shared__ float As[2][GBM * GBK];
  __shared__ float Bs[2][GBK * GBN];

  const int tid  = threadIdx.x;
  const int wave = tid >> 5;
  const int lane = tid & 31;
  const int l15  = lane & 15;
  const int kh   = lane >> 4;        // 0: K0/K1 half, 1: K2/K3 half
  const int wm   = wave >> 2;        // 0..1
  const int wn   = wave & 3;         // 0..3
  const int m0   = blockIdx.y * GBM;
  const int n0   = blockIdx.x * GBN;

  auto load_tile = [&](int buf, int k0) {
    const int r  = tid >> 2, c = (tid & 3) * 4;        // A: 64x16
    const int rb = tid >> 5, cb = (tid & 31) * 4;      // B: 16x128
    const float* gA  = A + (size_t)(m0 + r) * K + k0 + c;
    const float* gB0 = B + (size_t)(k0 + rb) * N + n0 + cb;
    const float* gB1 = B + (size_t)(k0 + rb + 8) * N + n0 + cb;
#ifdef HAVE_ASYNC_LDS
    ASYNC_COPY_B128(gA,  &As[buf][r * GBK + c]);
    ASYNC_COPY_B128(gB0, &Bs[buf][rb * GBN + cb]);
    ASYNC_COPY_B128(gB1, &Bs[buf][(rb + 8) * GBN + cb]);
#else
    *reinterpret_cast<float4*>(&As[buf][r * GBK + c]) =
        *reinterpret_cast<const float4*>(gA);
    *reinterpret_cast<float4*>(&Bs[buf][rb * GBN + cb]) =
        *reinterpret_cast<const float4*>(gB0);
    *reinterpret_cast<float4*>(&Bs[buf][(rb + 8) * GBN + cb]) =
        *reinterpret_cast<const float4*>(gB1);
#endif
  };

  v8f acc00 = v8f_zero(), acc01 = v8f_zero();
  v8f acc10 = v8f_zero(), acc11 = v8f_zero();

  load_tile(0, 0);
  int buf = 0;
  for (int k0 = 0; k0 < K; k0 += GBK) {
    if (k0 + GBK < K) {
      load_tile(buf ^ 1, k0 + GBK);   // prefetch next tile into other buffer
#ifdef HAVE_ASYNC_LDS
      WAIT_ASYNC(3);                  // current tile landed; next in flight
#endif
    } else {
#ifdef HAVE_ASYNC_LDS
      WAIT_ASYNC(0);
#endif
    }
    __syncthreads();

    const float* as = &As[buf][0];
    const float* bs = &Bs[buf][0];
#pragma unroll
    for (int kk = 0; kk < GBK; kk += 4) {
      const int ka = kk + 2 * kh;
      v2f a0 = {as[(wm * 32 + l15) * GBK + ka],
                as[(wm * 32 + l15) * GBK + ka + 1]};
      v2f a1 = {as[(wm * 32 + 16 + l15) * GBK + ka],
                as[(wm * 32 + 16 + l15) * GBK + ka + 1]};
      v2f b0 = {bs[ka * GBN + wn * 32 + l15],
                bs[(ka + 1) * GBN + wn * 32 + l15]};
      v2f b1 = {bs[ka * GBN + wn * 32 + 16 + l15],
                bs[(ka + 1) * GBN + wn * 32 + 16 + l15]};
      acc00 = wmma4(a0, b0, acc00);
      acc01 = wmma4(a0, b1, acc01);
      acc10 = wmma4(a1, b0, acc10);
      acc11 = wmma4(a1, b1, acc11);
    }
    __syncthreads();
    buf ^= 1;
  }

  // Epilogue: D layout -> row = tileM + r + 8*kh, col = tileN + l15
  v8f* accs[2][2] = {{&acc00, &acc01}, {&acc10, &acc11}};
#pragma unroll
  for (int ti = 0; ti < 2; ++ti) {
#pragma unroll
    for (int tj = 0; tj < 2; ++tj) {
      const int gn = n0 + wn * 32 + tj * 16 + l15;
      const float bb = bias ? bias[gn] : 0.0f;
      v8f d = *accs[ti][tj];
#pragma unroll
      for (int r = 0; r < 8; ++r) {
        const int gm = m0 + wm * 32 + ti * 16 + r + 8 * kh;
        C[(size_t)gm * N + gn] = d[r] + bb;
      }
    }
  }
}

// ---------------------------------------------------------------------------
// b12[s] = sum_k b_emb2state[k] * W_input_influence[k, s]   (256 wide)
// ---------------------------------------------------------------------------
__global__ void fuse_bias_kernel(const float* __restrict__ bvec,
                                 const float* __restrict__ W,  // [256,256]
                                 float* __restrict__ out) {
  const int s = blockIdx.x * blockDim.x + threadIdx.x;
  float acc = 0.f;
  for (int k = 0; k < 256; ++k) acc += bvec[k] * W[k * 256 + s];
  out[s] = acc;
}

// ---------------------------------------------------------------------------
// Pack weights so each lane's (k, k+1) B-fragment pair is one aligned float2:
//   Wp[((s*2 + kh)*N + n)*2 + {0,1}] = W[(4s+2kh)*N + n], W[(4s+2kh+1)*N + n]
// Turns 2 scattered b32 loads per WMMA in the serial scan into 1 b64 load.
// ---------------------------------------------------------------------------
__global__ void pack_pairs_kernel(const float* __restrict__ W,
                                  float* __restrict__ Wp, int N) {
  const int idx = blockIdx.x * blockDim.x + threadIdx.x;  // [0, 128*N)
  const int n = idx % N, s2 = idx / N;
  const int krow = 4 * (s2 >> 1) + 2 * (s2 & 1);
  Wp[idx * 2 + 0] = W[krow * N + n];
  Wp[idx * 2 + 1] = W[(krow + 1) * N + n];
}

// ---------------------------------------------------------------------------
// Sequential scan. One workgroup, 512 threads = 16 wave32s.
//   h (8x256, zero-padded to 16 rows) lives in LDS.
//   GEMM1: ns = h_pad @ Wsc + U[t]     (16 tiles, 1/wave, 2 acc chains)
//   GEMM2: g  = ns_pad @ Wglu + b_glu  (32 tiles, 2/wave, 4 acc chains)
//   GLU+LN: waves 0-7 (one per batch row) with shfl_xor reductions.
//   Waves 8-15 prefetch U[t+1] during the LN phase.
// Packed weights (768KB total) stay L2-hot across all 2048 steps.
// ---------------------------------------------------------------------------
__global__ __launch_bounds__(512) void scan_kernel(
    const float* __restrict__ U,      // [8, T, 256]
    const float* __restrict__ WscP,   // packed [128,256,2]
    const float* __restrict__ WgluP,  // packed [128,512,2]
    const float* __restrict__ bglu,   // [512]
    const float* __restrict__ gamma,  // [256]
    const float* __restrict__ beta,   // [256]
    float* __restrict__ HS,           // [8, T, 256]
    int T) {
  __shared__ float sh[16 * 256];   // padded h
  __shared__ float sns[16 * 256];  // padded ns
  __shared__ float sg[8 * 512];    // g (rows 0-7 only)

  const int tid  = threadIdx.x;
  const int wave = tid >> 5;
  const int lane = tid & 31;
  const int l15  = lane & 15;
  const int kh   = lane >> 4;

  for (int i = tid; i < 16 * 256; i += 512) { sh[i] = 0.f; sns[i] = 0.f; }
  __syncthreads();

  const float2* WscP2  = reinterpret_cast<const float2*>(WscP);
  const float2* WgluP2 = reinterpret_cast<const float2*>(WgluP);

  for (int t = 0; t < T; ++t) {
    // ---- GEMM1: ns = h_pad @ Wsc  (+ U[t] on rows 0..7) -------------------
    {
      const int n1 = wave * 16 + l15;  // output column
      v8f acc_e = v8f_zero(), acc_o = v8f_zero();
#pragma unroll 8
      for (int s = 0; s < 64; ++s) {
        const int ka = 4 * s + 2 * kh;
        v2f a = {sh[l15 * 256 + ka], sh[l15 * 256 + ka + 1]};
        const float2 bp = WscP2[(size_t)(s * 2 + kh) * 256 + n1];
        v2f b = {bp.x, bp.y};
        if (s & 1) acc_o = wmma4(a, b, acc_o);
        else       acc_e = wmma4(a, b, acc_e);
      }
      v8f acc = acc_e + acc_o;
      if (kh == 0) {  // rows 0..7 get U[b=row, t, n1]
#pragma unroll
        for (int r = 0; r < 8; ++r)
          acc[r] += U[((size_t)r * T + t) * 256 + n1];
      }
#pragma unroll
      for (int r = 0; r < 8; ++r)
        sns[(r + 8 * kh) * 256 + n1] = acc[r];  // rows 8..15 are exact zeros
    }
    __syncthreads();

    // ---- GEMM2: g = ns_pad @ Wglu + b_glu ---------------------------------
    {
      const int n2a = wave * 32 + l15;
      const int n2b = n2a + 16;
      v8f a0e = v8f_zero(), a0o = v8f_zero();
      v8f a1e = v8f_zero(), a1o = v8f_zero();
#pragma unroll 8
      for (int s = 0; s < 64; ++s) {
        const int ka = 4 * s + 2 * kh;
        v2f a = {sns[l15 * 256 + ka], sns[l15 * 256 + ka + 1]};
        const float2 bp0 = WgluP2[(size_t)(s * 2 + kh) * 512 + n2a];
        const float2 bp1 = WgluP2[(size_t)(s * 2 + kh) * 512 + n2b];
        v2f b0 = {bp0.x, bp0.y};
        v2f b1 = {bp1.x, bp1.y};
        if (s & 1) { a0o = wmma4(a, b0, a0o); a1o = wmma4(a, b1, a1o); }
        else       { a0e = wmma4(a, b0, a0e); a1e = wmma4(a, b1, a1e); }
      }
      v8f acc0 = a0e + a0o;
      v8f acc1 = a1e + a1o;
      if (kh == 0) {
        const float bb0 = bglu[n2a], bb1 = bglu[n2b];
#pragma unroll
        for (int r = 0; r < 8; ++r) {
          sg[r * 512 + n2a] = acc0[r] + bb0;
          sg[r * 512 + n2b] = acc1[r] + bb1;
        }
      }
    }
    __syncthreads();

    // ---- GLU + LayerNorm --------------------------------------------------
    if (wave < 8) {  // wave == batch row
      float vals[8];
      float s1 = 0.f, s2 = 0.f;
#pragma unroll
      for (int i = 0; i < 8; ++i) {
        const int j = lane + 32 * i;
        const float av = sg[wave * 512 + j];
        const float bv = sg[wave * 512 + 256 + j];
        const float v = av * (1.0f / (1.0f + __expf(-bv)));
        vals[i] = v;
        s1 += v;
        s2 += v * v;
      }
#pragma unroll
      for (int off = 16; off > 0; off >>= 1) {
        s1 += __shfl_xor(s1, off, 32);
        s2 += __shfl_xor(s2, off, 32);
      }
      const float mu  = s1 * (1.0f / 256.0f);
      const float var = s2 * (1.0f / 256.0f) - mu * mu;
      const float rs  = rsqrtf(var + 1e-5f);
#pragma unroll
      for (int i = 0; i < 8; ++i) {
        const int j = lane + 32 * i;
        const float hn = (vals[i] - mu) * rs * gamma[j] + beta[j];
        sh[wave * 256 + j] = hn;
        HS[((size_t)wave * T + t) * 256 + j] = hn;
      }
    } else {
      // idle waves warm the cache for the next timestep's U rows
      if (t + 1 < T) {
        const int r = wave - 8;  // 0..7
        __builtin_prefetch(&U[((size_t)r * T + (t + 1)) * 256 + lane * 8], 0, 0);
      }
    }
    __syncthreads();
  }
}

// ---------------------------------------------------------------------------
extern "C" void kernel_launch(void* const* d_in, const int* in_sizes, int n_in,
                              void* d_out, int out_size, void* d_ws,
                              size_t ws_size, hipStream_t stream) {
  const float* emb   = (const float*)d_in[0];   // [8,2048,1024]
  const float* Wsc   = (const float*)d_in[1];   // [256,256]
  const float* Winf  = (const float*)d_in[2];   // [256,256]
  const float* Wos   = (const float*)d_in[3];   // [256,256]
  const float* Wglu  = (const float*)d_in[4];   // [256,512]
  const float* bglu  = (const float*)d_in[5];   // [512]
  const float* gamma = (const float*)d_in[6];   // [256]
  const float* beta  = (const float*)d_in[7];   // [256]
  const float* We2s  = (const float*)d_in[8];   // [1024,256]
  const float* be2s  = (const float*)d_in[9];   // [256]
  const float* Ws2o  = (const float*)d_in[10];  // [256,1024]
  const float* bs2o  = (const float*)d_in[11];  // [1024]
  float* out = (float*)d_out;

  const int E = 1024, Bb = 8;
  const int M = in_sizes[0] / E;  // B*T = 16384
  const int T = M / Bb;           // 2048

  float* ws    = (float*)d_ws;
  float* W12   = ws;  ws += 1024 * 256;
  float* b12   = ws;  ws += 256;
  float* W34   = ws;  ws += 256 * 1024;
  float* WscP  = ws;  ws += 256 * 256;
  float* WgluP = ws;  ws += 256 * 512;
  float* Ubuf  = ws;  ws += (size_t)M * 256;
  float* HSb   = ws;  ws += (size_t)M * 256;

  // 1) fused weights / bias / packed scan weights
  fuse_bias_kernel<<<1, 256, 0, stream>>>(be2s, Winf, b12);
  pack_pairs_kernel<<<(128 * 256) / 256, 256, 0, stream>>>(Wsc, WscP, 256);
  pack_pairs_kernel<<<(128 * 512) / 256, 256, 0, stream>>>(Wglu, WgluP, 512);
  gemm_wmma_f32<<<dim3(256 / GBN, 1024 / GBM), 256, 0, stream>>>(
      We2s, Winf, nullptr, W12, 1024, 256, 256);
  gemm_wmma_f32<<<dim3(1024 / GBN, 256 / GBM), 256, 0, stream>>>(
      Wos, Ws2o, nullptr, W34, 256, 1024, 256);

  // 2) U = emb @ W12 + b12   [M,256]
  gemm_wmma_f32<<<dim3(256 / GBN, M / GBM), 256, 0, stream>>>(
      emb, W12, b12, Ubuf, M, 256, 1024);

  // 3) sequential recurrence
  scan_kernel<<<1, 512, 0, stream>>>(Ubuf, WscP, WgluP, bglu, gamma, beta, HSb,
                                     T);

  // 4) out = HS @ W34 + b_state2out   [M,1024]
  gemm_wmma_f32<<<dim3(1024 / GBN, M / GBM), 256, 0, stream>>>(
      HSb, W34, bs2o, out, M, 1024, 256);
}